// PositionalWordEmbedding_19645180412331
// MI455X (gfx1250) — compile-verified
//
#include <hip/hip_runtime.h>
#include <stdint.h>

// Problem constants from the reference: x[32,4096] int tokens, W[50257,256] f32.
#define BATCH 32
#define SEQ   4096
#define DIM   256
#define DIM4  (DIM / 4)   // 64 float4 per row

// Native clang vector type (required by __builtin_nontemporal_store, and
// lowers to b128 global loads/stores).
typedef float v4f __attribute__((ext_vector_type(4)));

// One block per sequence position s.
//   - wave 0 async-stages the 32 token ids x[:,s] into LDS (gfx1250
//     GLOBAL_LOAD_ASYNC_TO_LDS_B32, tracked by ASYNCcnt).
//   - each thread computes its 4 PE values with HW transcendentals
//     (1x v_exp_f32, 2x v_sin_f32, 2x v_cos_f32), reused for all 32 batches.
//   - gather W rows (L2-resident: 51.5 MB < 192 MB L2) with b128 loads,
//     add PE, nontemporal b128 store so the 128 MiB output stream does not
//     evict W from L2; prefetch the next batch's row via global_prefetch_b8.
__global__ __launch_bounds__(256) void pos_word_embed_kernel(
    const int* __restrict__ x,      // [BATCH, SEQ]
    const float* __restrict__ W,    // [VOCAB, DIM]
    float* __restrict__ out)        // [BATCH, SEQ, DIM]
{
    const int s    = blockIdx.x;        // sequence position
    const int tid  = threadIdx.x;       // 0..255 (8 wave32 waves)
    const int k    = tid & 63;          // float4 index along DIM
    const int bsub = tid >> 6;          // 0..3 batch sub-lane

    __shared__ int tok[BATCH];

    // --- async-stage the 32 strided token ids for this position into LDS ---
    if (tid < BATCH) {                  // exactly wave 0, EXEC all-ones
        const int* gp       = x + (size_t)tid * SEQ + s;
        uint32_t   lds_off  = (uint32_t)(uintptr_t)(&tok[tid]); // low 32b of generic LDS addr = LDS byte offset
        uint64_t   gaddr    = (uint64_t)(uintptr_t)gp;
        asm volatile("global_load_async_to_lds_b32 %0, %1, off"
                     :
                     : "v"(lds_off), "v"(gaddr)
                     : "memory");
        asm volatile("s_wait_asynccnt 0" ::: "memory");
    }
    __syncthreads();

    // --- sinusoidal PE for this thread's 4 lanes of D, computed once ---
    // pe[s,d] = sin(s * 10000^-(d/DIM))      d even
    //         = cos(s * 10000^-((d-1)/DIM))  d odd   (same freq as the even mate)
    // invf(d+2) = invf(d) * 10000^(-2/DIM): fold the 2nd exp into a constant mul.
    const float ln1e4_over_dim = 9.210340371976184f / (float)DIM; // ln(10000)/DIM
    const float freq_step      = 0.93057204f;                     // 10000^(-2/256)
    const float fs    = (float)s;
    const float invf0 = __expf(-(float)(4 * k) * ln1e4_over_dim); // v_exp_f32
    const float a0 = fs * invf0;
    const float a1 = a0 * freq_step;
    v4f pe;
    pe.x = __sinf(a0);   // v_sin_f32
    pe.y = __cosf(a0);   // v_cos_f32
    pe.z = __sinf(a1);
    pe.w = __cosf(a1);

    const v4f* __restrict__ Wv = (const v4f*)W;
    v4f*       __restrict__ Ov = (v4f*)out;

    // --- gather + add + NT store, 8 batches per thread, prefetch next row ---
    int b     = bsub;
    int token = tok[b];
    #pragma unroll
    for (int it = 0; it < BATCH / 4; ++it, b += 4) {
        const int tnext = (b + 4 < BATCH) ? tok[b + 4] : token;
        // gfx1250 global_prefetch_b8: warm the next gathered row (L2-resident W)
        __builtin_prefetch(&Wv[(size_t)tnext * DIM4 + k], 0, 3);

        const v4f w = Wv[(size_t)token * DIM4 + k];      // global_load_b128
        const v4f r = w + pe;
        // Nontemporal b128 store: stream the write-once output past L2 so the
        // embedding table stays resident for the gather.
        __builtin_nontemporal_store(r, &Ov[((size_t)b * SEQ + s) * DIM4 + k]);

        token = tnext;
    }
}

extern "C" void kernel_launch(void* const* d_in, const int* in_sizes, int n_in,
                              void* d_out, int out_size, void* d_ws, size_t ws_size,
                              hipStream_t stream) {
    const int*   x = (const int*)d_in[0];    // token ids [32,4096]
    const float* W = (const float*)d_in[1];  // embedding table [50257,256]
    float*     out = (float*)d_out;          // [32,4096,256]
    (void)in_sizes; (void)n_in; (void)out_size; (void)d_ws; (void)ws_size;

    pos_word_embed_kernel<<<SEQ, 256, 0, stream>>>(x, W, out);
}